// GATv2ActorCritic_44710609552119
// MI455X (gfx1250) — compile-verified
//
#include <hip/hip_runtime.h>
#include <math.h>

typedef __attribute__((ext_vector_type(2))) float v2f;
typedef __attribute__((ext_vector_type(8))) float v8f;

#define NNODES   100000
#define NEDGES   1600000
#define NGRAPH   16
#define IN_CH    32
#define CH       32      // per-head channels (HID == OUT_CH == 32)
#define NHEADS   4
#define EDIM     16
#define DFEAT    128     // NHEADS*CH
#define NEG_SLOPE 0.2f
#define NOUT_HEAD 25     // 15 gmm + 10 proposals
#define LPAD     4       // LDS row padding (floats) -> lane stride = 4 banks, conflict-free

// ---------- float atomic-max via monotone unsigned encoding ----------
__device__ __forceinline__ unsigned f2ord(float f) {
  unsigned b = __float_as_uint(f);
  return (b & 0x80000000u) ? ~b : (b | 0x80000000u);
}
__device__ __forceinline__ float ord2f(unsigned u) {
  unsigned b = (u & 0x80000000u) ? (u & 0x7FFFFFFFu) : ~u;
  return __uint_as_float(b);
}

// ---------- WMMA f32 GEMM: C[M,Nout] = A[M,K] @ W[Nout,K]^T + bias, optional relu
// Block: 256 threads = 8 waves; one 16-row M-tile per block; wave w owns col tile w.
// A tile (16xK) and full W (NoutxK) staged in LDS with coalesced float4 loads, so the
// WMMA loop is ds_load_b64 + v_wmma with no global-latency stalls.
// Fragment layout per ISA 7.12.2:
//   A 16x4: lanes 0-15 hold M=lane, K={k,k+1}; lanes 16-31 K={k+2,k+3}.
//   B 4x16: lanes hold N=lane%16, same K split.
//   D: vgpr r -> row = r + (lane/16)*8, col = lane%16.
__global__ void gemm_bias_wmma(const float* __restrict__ A,
                               const float* __restrict__ W,
                               const float* __restrict__ bias,
                               float* __restrict__ C,
                               int M, int K, int Nout, int act) {
  extern __shared__ float smem[];
  const int Kp = K + LPAD;
  float* sA = smem;            // [16 * Kp]
  float* sW = smem + 16 * Kp;  // [Nout * Kp]

  const int tid  = threadIdx.x;
  const int wave = tid >> 5;
  const int lane = tid & 31;
  const int half = lane >> 4;
  const int lm   = lane & 15;
  const int rbase = blockIdx.x * 16;
  const int kv = K >> 2;       // float4s per row

  // stage A tile (coalesced)
  for (int i = tid; i < 16 * kv; i += blockDim.x) {
    int row = i / kv, cv = i - row * kv;
    int arow = rbase + row; if (arow >= M) arow = M - 1;
    float4 v = *(const float4*)(A + (size_t)arow * K + cv * 4);
    float* p = &sA[row * Kp + cv * 4];
    p[0] = v.x; p[1] = v.y; p[2] = v.z; p[3] = v.w;
  }
  // stage W (coalesced)
  for (int i = tid; i < Nout * kv; i += blockDim.x) {
    int row = i / kv, cv = i - row * kv;
    float4 v = *(const float4*)(W + (size_t)row * K + cv * 4);
    float* p = &sW[row * Kp + cv * 4];
    p[0] = v.x; p[1] = v.y; p[2] = v.z; p[3] = v.w;
  }
  __syncthreads();

  const int nTiles = Nout >> 4;
  const int nw = blockDim.x >> 5;
  for (int nt = wave; nt < nTiles; nt += nw) {
    const int cbase = nt * 16;
    const float* abase = &sA[lm * Kp + half * 2];
    const float* wbase = &sW[(cbase + lm) * Kp + half * 2];
    v8f acc = {};
    #pragma unroll 8
    for (int k = 0; k < K; k += 4) {
      v2f a = *(const v2f*)(abase + k);
      v2f b = *(const v2f*)(wbase + k);
      acc = __builtin_amdgcn_wmma_f32_16x16x4_f32(false, a, false, b,
                                                  (short)0, acc, false, false);
    }
    float bv = bias ? bias[cbase + lm] : 0.0f;
    #pragma unroll
    for (int r = 0; r < 8; ++r) {
      int rr = rbase + r + half * 8;
      if (rr < M) {
        float v = acc[r] + bv;
        if (act == 1) v = v > 0.0f ? v : 0.0f;
        C[(size_t)rr * Nout + cbase + lm] = v;
      }
    }
  }
}

// ---------- self-loop edge_attr = mean of incoming edge attrs ----------
__global__ void k_scatter_ea(const int* __restrict__ dst,
                             const float* __restrict__ ea,
                             float* __restrict__ loop_sum,
                             float* __restrict__ cnt, int E) {
  long long idx = (long long)blockIdx.x * blockDim.x + threadIdx.x;
  if (idx >= (long long)E * EDIM) return;
  int e = (int)(idx >> 4);
  int j = (int)(idx & 15);
  int d = dst[e];
  atomicAdd(&loop_sum[(size_t)d * EDIM + j], ea[idx]);
  if (j == 0) atomicAdd(&cnt[d], 1.0f);
}

__global__ void k_finalize_loop(float* __restrict__ loop_attr,
                                const float* __restrict__ cnt, int Nn) {
  long long idx = (long long)blockIdx.x * blockDim.x + threadIdx.x;
  if (idx >= (long long)Nn * EDIM) return;
  int n = (int)(idx >> 4);
  loop_attr[idx] /= fmaxf(cnt[n], 1.0f);
}

// ---------- edge pass 1: attention logits + segment max ----------
__global__ void k_edge_logits(const float* __restrict__ xl,
                              const float* __restrict__ xr,
                              const float* __restrict__ We,   // [DFEAT, EDIM]
                              const float* __restrict__ att,  // [NHEADS, CH]
                              const float* __restrict__ ea,   // [E, EDIM]
                              const float* __restrict__ loop_attr,
                              const int* __restrict__ src,
                              const int* __restrict__ dst,
                              float* __restrict__ elog,       // [(E+N)*NHEADS]
                              unsigned* __restrict__ mx_enc,  // [N*NHEADS]
                              int E, int Nn) {
  __shared__ float sWe[DFEAT * EDIM];
  __shared__ float sAtt[DFEAT];
  for (int i = threadIdx.x; i < DFEAT * EDIM; i += blockDim.x) sWe[i] = We[i];
  for (int i = threadIdx.x; i < DFEAT; i += blockDim.x) sAtt[i] = att[i];
  __syncthreads();
  long long idx = (long long)blockIdx.x * blockDim.x + threadIdx.x;
  long long total = (long long)(E + Nn) * NHEADS;
  if (idx >= total) return;
  int e = (int)(idx >> 2);
  int h = (int)(idx & 3);
  int s, d;
  const float* eap;
  if (e < E) { s = src[e]; d = dst[e]; eap = ea + (size_t)e * EDIM; }
  else       { s = e - E;  d = s;      eap = loop_attr + (size_t)s * EDIM; }
  float eav[EDIM];
  #pragma unroll
  for (int j = 0; j < EDIM; ++j) eav[j] = eap[j];
  const float* xls = xl + (size_t)s * DFEAT + h * CH;
  const float* xrd = xr + (size_t)d * DFEAT + h * CH;
  float logit = 0.0f;
  #pragma unroll 4
  for (int c = 0; c < CH; ++c) {
    float eec = 0.0f;
    const float* wrow = &sWe[(h * CH + c) * EDIM];
    #pragma unroll
    for (int j = 0; j < EDIM; ++j) eec += eav[j] * wrow[j];
    float m = xls[c] + xrd[c] + eec;
    m = m > 0.0f ? m : NEG_SLOPE * m;
    logit += sAtt[h * CH + c] * m;
  }
  elog[idx] = logit;
  atomicMax(&mx_enc[(size_t)d * NHEADS + h], f2ord(logit));
}

// ---------- edge pass 2: exp(logit - max), segment sum ----------
__global__ void k_edge_exp(const int* __restrict__ dst,
                           const unsigned* __restrict__ mx_enc,
                           float* __restrict__ elog,
                           float* __restrict__ den,
                           int E, int Nn) {
  long long idx = (long long)blockIdx.x * blockDim.x + threadIdx.x;
  if (idx >= (long long)(E + Nn) * NHEADS) return;
  int e = (int)(idx >> 2);
  int h = (int)(idx & 3);
  int d = (e < E) ? dst[e] : (e - E);
  float mxv = ord2f(mx_enc[(size_t)d * NHEADS + h]);
  float ex = expf(elog[idx] - mxv);
  elog[idx] = ex;
  atomicAdd(&den[(size_t)d * NHEADS + h], ex);
}

// ---------- edge pass 3: out[d] += alpha * xl[s]  (4 channels per thread) ----------
__global__ void k_edge_aggregate(const int* __restrict__ src,
                                 const int* __restrict__ dst,
                                 const float* __restrict__ elog,
                                 const float* __restrict__ den,
                                 const float* __restrict__ xl,
                                 float* __restrict__ out,
                                 int E, int Nn) {
  long long idx = (long long)blockIdx.x * blockDim.x + threadIdx.x;
  if (idx >= (long long)(E + Nn) * 32) return;
  int e = (int)(idx >> 5);
  int q = (int)(idx & 31);          // channels q*4 .. q*4+3
  int h = q >> 3;
  int s, d;
  if (e < E) { s = src[e]; d = dst[e]; } else { s = e - E; d = s; }
  float alpha = elog[(size_t)e * NHEADS + h] / den[(size_t)d * NHEADS + h];
  const float4 xv = *(const float4*)(xl + (size_t)s * DFEAT + q * 4);
  float* op = out + (size_t)d * DFEAT + q * 4;
  atomicAdd(op + 0, alpha * xv.x);
  atomicAdd(op + 1, alpha * xv.y);
  atomicAdd(op + 2, alpha * xv.z);
  atomicAdd(op + 3, alpha * xv.w);
}

// ---------- y = elu(acc + bias) in place ----------
__global__ void k_bias_elu(float* __restrict__ y, const float* __restrict__ bias,
                           long long total) {
  long long idx = (long long)blockIdx.x * blockDim.x + threadIdx.x;
  if (idx >= total) return;
  int c = (int)(idx & (DFEAT - 1));
  float v = y[idx] + bias[c];
  y[idx] = v > 0.0f ? v : expm1f(v);
}

// ---------- pooled[b] += (acc[n] + bias2), pcnt[b] += 1 ----------
__global__ void k_pool(const float* __restrict__ acc, const float* __restrict__ bias2,
                       const int* __restrict__ batch,
                       float* __restrict__ pooled, float* __restrict__ pcnt, int Nn) {
  long long idx = (long long)blockIdx.x * blockDim.x + threadIdx.x;
  if (idx >= (long long)Nn * DFEAT) return;
  int n = (int)(idx >> 7);
  int c = (int)(idx & (DFEAT - 1));
  int b = batch[n];
  atomicAdd(&pooled[(size_t)b * DFEAT + c], acc[idx] + bias2[c]);
  if (c == 0) atomicAdd(&pcnt[b], 1.0f);
}

__global__ void k_gmean(const float* __restrict__ pooled, const float* __restrict__ pcnt,
                        float* __restrict__ g) {
  int idx = blockIdx.x * blockDim.x + threadIdx.x;
  if (idx >= NGRAPH * DFEAT) return;
  g[idx] = pooled[idx] / fmaxf(pcnt[idx >> 7], 1.0f);
}

// ---------- final heads: [B,15] gmm (from relu hidden) ++ [B,10] proposals ----------
__global__ void k_heads(const float* __restrict__ hbuf, const float* __restrict__ g,
                        const float* __restrict__ Wg2, const float* __restrict__ bg2,
                        const float* __restrict__ Wp,  const float* __restrict__ bp,
                        float* __restrict__ out) {
  int idx = blockIdx.x * blockDim.x + threadIdx.x;
  if (idx >= NGRAPH * NOUT_HEAD) return;
  int b = idx / NOUT_HEAD;
  int j = idx % NOUT_HEAD;
  float acc;
  if (j < 15) {
    acc = bg2[j];
    const float* h = hbuf + (size_t)b * DFEAT;
    const float* w = Wg2 + (size_t)j * DFEAT;
    for (int k = 0; k < DFEAT; ++k) acc += h[k] * w[k];
  } else {
    int jj = j - 15;
    acc = bp[jj];
    const float* gv = g + (size_t)b * DFEAT;
    const float* w = Wp + (size_t)jj * DFEAT;
    for (int k = 0; k < DFEAT; ++k) acc += gv[k] * w[k];
  }
  out[idx] = acc;
}

static inline unsigned gsz(long long total, int block) {
  return (unsigned)((total + block - 1) / block);
}

static inline size_t gemm_lds_bytes(int K, int Nout) {
  return (size_t)(16 + Nout) * (K + LPAD) * sizeof(float);
}

extern "C" void kernel_launch(void* const* d_in, const int* in_sizes, int n_in,
                              void* d_out, int out_size, void* d_ws, size_t ws_size,
                              hipStream_t stream) {
  const float* x         = (const float*)d_in[0];
  const int*   edge_idx  = (const int*)d_in[1];
  const float* edge_attr = (const float*)d_in[2];
  const int*   batch     = (const int*)d_in[3];
  const float* W1l = (const float*)d_in[4];  const float* b1l  = (const float*)d_in[5];
  const float* W1r = (const float*)d_in[6];  const float* b1r  = (const float*)d_in[7];
  const float* W1e = (const float*)d_in[8];  const float* att1 = (const float*)d_in[9];
  const float* bias1 = (const float*)d_in[10];
  const float* W2l = (const float*)d_in[11]; const float* b2l  = (const float*)d_in[12];
  const float* W2r = (const float*)d_in[13]; const float* b2r  = (const float*)d_in[14];
  const float* W2e = (const float*)d_in[15]; const float* att2 = (const float*)d_in[16];
  const float* bias2 = (const float*)d_in[17];
  const float* Wg1 = (const float*)d_in[18]; const float* bg1 = (const float*)d_in[19];
  const float* Wg2 = (const float*)d_in[20]; const float* bg2 = (const float*)d_in[21];
  const float* Wp  = (const float*)d_in[22]; const float* bp  = (const float*)d_in[23];

  const int Nn = in_sizes[0] / IN_CH;     // 100000
  const int E  = in_sizes[1] / 2;         // 1600000
  const int* src = edge_idx;
  const int* dst = edge_idx + E;

  // ---- workspace layout (float elements) ----
  float* ws = (float*)d_ws;
  size_t o = 0;
  float*    loopA  = ws + o; o += (size_t)Nn * EDIM;        // self-loop attrs
  float*    cnt    = ws + o; o += (size_t)Nn;
  float*    bufA   = ws + o; o += (size_t)Nn * DFEAT;       // xl
  float*    bufB   = ws + o; o += (size_t)Nn * DFEAT;       // xr
  float*    bufC   = ws + o; o += (size_t)Nn * DFEAT;       // layer1 out (y) / layer2 out
  float*    elog   = ws + o; o += (size_t)(E + Nn) * NHEADS;
  unsigned* mx_enc = (unsigned*)(ws + o); o += (size_t)Nn * NHEADS;
  float*    den    = ws + o; o += (size_t)Nn * NHEADS;
  float*    pooled = ws + o; o += (size_t)NGRAPH * DFEAT;
  float*    pcnt   = ws + o; o += (size_t)NGRAPH;
  float*    gbuf   = ws + o; o += (size_t)NGRAPH * DFEAT;
  float*    hbuf   = ws + o; o += (size_t)NGRAPH * DFEAT;

  const int BLK = 256;
  const long long edH = (long long)(E + Nn) * NHEADS;
  const long long edC = (long long)(E + Nn) * 32;
  const long long nf  = (long long)Nn * DFEAT;
  const size_t lds32  = gemm_lds_bytes(IN_CH, DFEAT);
  const size_t lds128 = gemm_lds_bytes(DFEAT, DFEAT);

  // ---- self-loop edge attrs (shared by both layers) ----
  hipMemsetAsync(loopA, 0, sizeof(float) * (size_t)Nn * EDIM, stream);
  hipMemsetAsync(cnt,   0, sizeof(float) * (size_t)Nn, stream);
  k_scatter_ea<<<gsz((long long)E * EDIM, BLK), BLK, 0, stream>>>(dst, edge_attr, loopA, cnt, E);
  k_finalize_loop<<<gsz((long long)Nn * EDIM, BLK), BLK, 0, stream>>>(loopA, cnt, Nn);

  // ================= layer 1 =================
  gemm_bias_wmma<<<(Nn + 15) / 16, BLK, lds32, stream>>>(x, W1l, b1l, bufA, Nn, IN_CH, DFEAT, 0);
  gemm_bias_wmma<<<(Nn + 15) / 16, BLK, lds32, stream>>>(x, W1r, b1r, bufB, Nn, IN_CH, DFEAT, 0);
  hipMemsetAsync(mx_enc, 0, sizeof(unsigned) * (size_t)Nn * NHEADS, stream);
  hipMemsetAsync(den,    0, sizeof(float) * (size_t)Nn * NHEADS, stream);
  k_edge_logits<<<gsz(edH, BLK), BLK, 0, stream>>>(bufA, bufB, W1e, att1, edge_attr,
                                                   loopA, src, dst, elog, mx_enc, E, Nn);
  k_edge_exp<<<gsz(edH, BLK), BLK, 0, stream>>>(dst, mx_enc, elog, den, E, Nn);
  hipMemsetAsync(bufC, 0, sizeof(float) * nf, stream);
  k_edge_aggregate<<<gsz(edC, BLK), BLK, 0, stream>>>(src, dst, elog, den, bufA, bufC, E, Nn);
  k_bias_elu<<<gsz(nf, BLK), BLK, 0, stream>>>(bufC, bias1, nf);

  // ================= layer 2 =================
  gemm_bias_wmma<<<(Nn + 15) / 16, BLK, lds128, stream>>>(bufC, W2l, b2l, bufA, Nn, DFEAT, DFEAT, 0);
  gemm_bias_wmma<<<(Nn + 15) / 16, BLK, lds128, stream>>>(bufC, W2r, b2r, bufB, Nn, DFEAT, DFEAT, 0);
  hipMemsetAsync(mx_enc, 0, sizeof(unsigned) * (size_t)Nn * NHEADS, stream);
  hipMemsetAsync(den,    0, sizeof(float) * (size_t)Nn * NHEADS, stream);
  k_edge_logits<<<gsz(edH, BLK), BLK, 0, stream>>>(bufA, bufB, W2e, att2, edge_attr,
                                                   loopA, src, dst, elog, mx_enc, E, Nn);
  k_edge_exp<<<gsz(edH, BLK), BLK, 0, stream>>>(dst, mx_enc, elog, den, E, Nn);
  hipMemsetAsync(bufC, 0, sizeof(float) * nf, stream);   // y already consumed by GEMMs
  k_edge_aggregate<<<gsz(edC, BLK), BLK, 0, stream>>>(src, dst, elog, den, bufA, bufC, E, Nn);

  // ================= pooling + heads =================
  hipMemsetAsync(pooled, 0, sizeof(float) * (size_t)NGRAPH * DFEAT, stream);
  hipMemsetAsync(pcnt,   0, sizeof(float) * (size_t)NGRAPH, stream);
  k_pool<<<gsz(nf, BLK), BLK, 0, stream>>>(bufC, bias2, batch, pooled, pcnt, Nn);
  k_gmean<<<gsz(NGRAPH * DFEAT, BLK), BLK, 0, stream>>>(pooled, pcnt, gbuf);
  gemm_bias_wmma<<<1, BLK, lds128, stream>>>(gbuf, Wg1, bg1, hbuf, NGRAPH, DFEAT, DFEAT, 1);
  k_heads<<<gsz(NGRAPH * NOUT_HEAD, BLK), BLK, 0, stream>>>(hbuf, gbuf, Wg2, bg2, Wp, bp,
                                                            (float*)d_out);
}